// VectorQuantizer_38457137168614
// MI455X (gfx1250) — compile-verified
//
#include <hip/hip_runtime.h>

typedef __attribute__((ext_vector_type(2))) float v2f;
typedef __attribute__((ext_vector_type(8))) float v8f;

#define KCODES   8192
#define DDIM     512
#define MROWS    16384            // 8 * 2048
#define MT       64               // rows per block
#define NT       32               // codes per iteration
#define PITCH    516              // 512 + 4 pad floats: pitch % 64 == 4 -> conflict-free frag reads
#define NITER    (KCODES / NT)    // 256
#define THREADS  256
#define Z_ELEMS  (MROWS * DDIM)   // 8388608
#define FLT_BIG  3.4028235e38f

// ---------------------------------------------------------------------------
// Kernel 1: e2[k] = sum_d embed[k][d]^2   (one wave per code row)
// ---------------------------------------------------------------------------
__global__ __launch_bounds__(256)
void vq_e2_kernel(const float* __restrict__ ew, float* __restrict__ e2) {
    const int lane = threadIdx.x & 31;
    const int wave = threadIdx.x >> 5;
    const int row  = blockIdx.x * 8 + wave;
    const float* p = ew + (size_t)row * DDIM + lane * 4;
    float s = 0.f;
#pragma unroll
    for (int i = 0; i < 4; ++i) {
        float4 v = *(const float4*)(p + i * 128);
        s += v.x * v.x + v.y * v.y + v.z * v.z + v.w * v.w;
    }
#pragma unroll
    for (int off = 16; off > 0; off >>= 1) s += __shfl_xor(s, off, 32);
    if (lane == 0) e2[row] = s;
}

// ---------------------------------------------------------------------------
// Kernel 2: main VQ kernel. One block owns 64 z-rows, sweeps all 8192 codes.
// dist = -2 * z.e + |e|^2  (|z|^2 dropped: constant per row for argmin)
// Embed tiles double-buffered in LDS via GLOBAL_LOAD_ASYNC_TO_LDS_B128:
// zero data VGPRs, transfers overlap the WMMA k-loop, ASYNCcnt-tracked.
// ---------------------------------------------------------------------------
__global__ __launch_bounds__(THREADS)
void vq_main_kernel(const float* __restrict__ z, const float* __restrict__ ew,
                    const float* __restrict__ e2, float* __restrict__ out,
                    float* __restrict__ out_codes, float* __restrict__ partials) {
    extern __shared__ float smem[];
    const int tid   = threadIdx.x;
    const int lane  = tid & 31;
    const int wave  = tid >> 5;
    const int mwave = wave >> 1;        // 0..3 : which 16-row stripe
    const int nwave = wave & 1;         // 0..1 : which 16-col stripe
    const int l16   = lane & 15;
    const int khalf = (lane >> 4) << 1; // 0 or 2 : K offset within frag
    const int grow  = blockIdx.x * MT;

    // LDS layout (float indices from `smem`; integer offsets keep everything
    // provably in the LDS address space -> ds_load, never flat)
    const int ZS    = 0;                      // 64 x 516 z tile
    const int ES    = ZS + MT * PITCH;        // 2 x (32 x 516) embed buffers
    const int CMIN  = ES + 2 * NT * PITCH;    // 64 x 2 floats
    const int CIDX  = CMIN + 2 * MT;          // 64 x 2 ints
    const int CODES = CIDX + 2 * MT;          // 64 ints
    const int RED   = CODES + MT;             // 256 floats

    float* zs      = smem + ZS;
    float* cminb   = smem + CMIN;
    int*   cidxb   = (int*)(smem + CIDX);
    int*   codes_s = (int*)(smem + CODES);
    float* red     = smem + RED;

    // ---- stage z tile (64 x 512) into LDS, b128 coalesced ----
    {
        const float* zsrc = z + (size_t)grow * DDIM;
#pragma unroll
        for (int i = 0; i < 32; ++i) {
            int f = tid + i * THREADS;
            int row = f >> 7;
            int col = (f & 127) << 2;
            float4 v = *(const float4*)(zsrc + row * DDIM + col);
            *(float4*)(zs + row * PITCH + col) = v;
        }
    }
    // ---- stage embed tile 0 into buffer 0 ----
    {
#pragma unroll
        for (int i = 0; i < 16; ++i) {
            int f = tid + i * THREADS;
            int row = f >> 7;
            int col = (f & 127) << 2;
            float4 v = *(const float4*)(ew + row * DDIM + col);
            *(float4*)(smem + ES + row * PITCH + col) = v;
        }
    }
    __syncthreads();

    float rmin[8];
    int   ridx[8];
#pragma unroll
    for (int r = 0; r < 8; ++r) { rmin[r] = FLT_BIG; ridx[r] = 0; }

    const float* Abase = zs + (mwave * 16 + l16) * PITCH + khalf;
    const int bfragoff = (nwave * 16 + l16) * PITCH + khalf;
    const int row2 = tid >> 7;            // this thread's base row (0/1) in tile copies
    const int col2 = (tid & 127) << 2;    // this thread's column (float index)
    int cbuf = 0;  // which embed buffer holds the current tile (0/1)

    for (int it = 0; it < NITER; ++it) {
        const int  nbase = it * NT;
        const int  mycol = nbase + nwave * 16 + l16;
        const bool more  = (it + 1 < NITER);

        // ---- async-prefetch next embed tile straight into the other buffer ----
        // global_load_async_to_lds_b128: vdst = LDS byte address, vaddr = 64-bit
        // global address. 16 issues x 256 threads = 64KB tile. No data VGPRs.
        if (more) {
            unsigned long long g =
                (unsigned long long)(ew + (size_t)(nbase + NT + row2) * DDIM + col2);
            unsigned ldsa = (unsigned)(uintptr_t)(smem + ES +
                                (cbuf ^ 1) * (NT * PITCH) + row2 * PITCH + col2);
#pragma unroll
            for (int i = 0; i < 16; ++i) {
                asm volatile("global_load_async_to_lds_b128 %0, %1, off"
                             :: "v"(ldsa), "v"(g) : "memory");
                g    += 2ull * DDIM * sizeof(float);   // +2 rows in global
                ldsa += 2u * PITCH * sizeof(float);    // +2 rows in LDS
            }
        }

        // ---- 16x16 f32 WMMA over K=512, chunked frag loads for clean clauses ----
        // Force the whole buffer offset into one VGPR so every ds_load gets a
        // small literal offset (enables ds_load_2addr pairing for B too).
        int boff = ES + cbuf * (NT * PITCH) + bfragoff;
        asm volatile("" : "+v"(boff));
        const float* Bbase = smem + boff;

        v8f acc0 = {0.f, 0.f, 0.f, 0.f, 0.f, 0.f, 0.f, 0.f};
        v8f acc1 = {0.f, 0.f, 0.f, 0.f, 0.f, 0.f, 0.f, 0.f};
#pragma unroll 1
        for (int tc = 0; tc < DDIM / 4; tc += 8) {   // 16 chunks of 8 k-steps
            v2f af[8], bf[8];
#pragma unroll
            for (int u = 0; u < 8; ++u) {
                af[u] = *(const v2f*)(Abase + 4 * (tc + u));
                bf[u] = *(const v2f*)(Bbase + 4 * (tc + u));
            }
#pragma unroll
            for (int u = 0; u < 8; u += 2) {
                acc0 = __builtin_amdgcn_wmma_f32_16x16x4_f32(
                    false, af[u], false, bf[u], (short)0, acc0, false, false);
                acc1 = __builtin_amdgcn_wmma_f32_16x16x4_f32(
                    false, af[u + 1], false, bf[u + 1], (short)0, acc1, false, false);
            }
        }

        // ---- per-lane running argmin update (pure VALU, branchless) ----
        const float e2v = e2[mycol];
#pragma unroll
        for (int r = 0; r < 8; ++r) {
            float v = fmaf(-2.f, acc0[r] + acc1[r], e2v);
            bool take = v < rmin[r];          // strict <: earliest index wins
            rmin[r] = take ? v : rmin[r];
            ridx[r] = take ? mycol : ridx[r];
        }

        // ---- wait for this wave's async tile transfers, then publish ----
        asm volatile("s_wait_asynccnt 0x0" ::: "memory");
        __syncthreads();
        cbuf ^= 1;
    }

    // ---- ONE cross-lane butterfly per row (16-lane halves), branchless ----
#pragma unroll
    for (int r = 0; r < 8; ++r) {
        float v  = rmin[r];
        int   ix = ridx[r];
#pragma unroll
        for (int off = 1; off < 16; off <<= 1) {
            float ov = __shfl_xor(v, off, 16);
            int   oi = __shfl_xor(ix, off, 16);
            bool take = (ov < v) || (ov == v && oi < ix);
            v  = take ? ov : v;
            ix = take ? oi : ix;
        }
        rmin[r] = v;
        ridx[r] = ix;
    }

    // ---- combine the two n-wave streams per row ----
    if (lane == 0 || lane == 16) {
        int rowoff = mwave * 16 + ((lane >> 4) << 3);  // +0 (lane 0) or +8 (lane 16)
#pragma unroll
        for (int r = 0; r < 8; ++r) {
            cminb[(rowoff + r) * 2 + nwave] = rmin[r];
            cidxb[(rowoff + r) * 2 + nwave] = ridx[r];
        }
    }
    __syncthreads();
    if (tid < MT) {
        float v0 = cminb[tid * 2 + 0]; int i0 = cidxb[tid * 2 + 0];
        float v1 = cminb[tid * 2 + 1]; int i1 = cidxb[tid * 2 + 1];
        int code = (v1 < v0 || (v1 == v0 && i1 < i0)) ? i1 : i0;
        codes_s[tid] = code;
        out_codes[grow + tid] = (float)code;
    }
    __syncthreads();

    // ---- phase 2: z_q gather, straight-through output, commit-loss partial ----
    float lsum = 0.f;
    for (int r = 0; r < MT; ++r) {
        const int c = codes_s[r];
        const float* eb = ew + (size_t)c * DDIM;
#pragma unroll
        for (int j = 0; j < DDIM / THREADS; ++j) {
            int d = tid + j * THREADS;
            float zv = zs[r * PITCH + d];
            float zq = eb[d];
            out[(size_t)(grow + r) * DDIM + d] = zv + (zq - zv);  // z + sg(z_q - z)
            float df = zv - zq;
            lsum = fmaf(df, df, lsum);
        }
    }
    red[tid] = lsum;
    __syncthreads();
#pragma unroll
    for (int s = THREADS / 2; s > 0; s >>= 1) {
        if (tid < s) red[tid] += red[tid + s];
        __syncthreads();
    }
    if (tid == 0) partials[blockIdx.x] = red[0];
}

// ---------------------------------------------------------------------------
// Kernel 3: deterministic final loss reduction (256 partials, fixed order)
// ---------------------------------------------------------------------------
__global__ __launch_bounds__(256)
void vq_loss_kernel(const float* __restrict__ partials, float* __restrict__ loss) {
    __shared__ float red[256];
    red[threadIdx.x] = partials[threadIdx.x];
    __syncthreads();
#pragma unroll
    for (int s = 128; s > 0; s >>= 1) {
        if (threadIdx.x < s) red[threadIdx.x] += red[threadIdx.x + s];
        __syncthreads();
    }
    if (threadIdx.x == 0) *loss = 0.1f * (red[0] / (float)Z_ELEMS);
}

// ---------------------------------------------------------------------------
extern "C" void kernel_launch(void* const* d_in, const int* in_sizes, int n_in,
                              void* d_out, int out_size, void* d_ws, size_t ws_size,
                              hipStream_t stream) {
    const float* z  = (const float*)d_in[0];   // (8, 2048, 512) fp32
    const float* ew = (const float*)d_in[1];   // (8192, 512) fp32

    float* out       = (float*)d_out;
    float* out_codes = out + (size_t)Z_ELEMS;  // 16384 codes (as float)
    float* out_loss  = out_codes + MROWS;      // 1 scalar

    float* e2       = (float*)d_ws;            // 8192 floats
    float* partials = e2 + KCODES;             // 256 floats

    vq_e2_kernel<<<KCODES / 8, 256, 0, stream>>>(ew, e2);

    const size_t smem_bytes =
        (size_t)(MT * PITCH + 2 * NT * PITCH) * sizeof(float)  // z tile + 2 embed bufs
        + 2 * MT * sizeof(float)                               // cmin
        + 2 * MT * sizeof(int)                                 // cidx
        + MT * sizeof(int)                                     // codes_s
        + THREADS * sizeof(float);                             // red
    hipFuncSetAttribute(reinterpret_cast<const void*>(vq_main_kernel),
                        hipFuncAttributeMaxDynamicSharedMemorySize, (int)smem_bytes);
    vq_main_kernel<<<MROWS / MT, THREADS, smem_bytes, stream>>>(z, ew, e2, out,
                                                               out_codes, partials);

    vq_loss_kernel<<<1, 256, 0, stream>>>(partials, out_loss);
}